// HeteroTopologyEncoder_26431228740059
// MI455X (gfx1250) — compile-verified
//
#include <hip/hip_runtime.h>
#include <math.h>

#define N_VEH    50000
#define N_RSU    500
#define E_V2V    800000
#define E_V2I    200000
#define E_I2V    200000
#define E_V2V_SL (E_V2V + N_VEH)   // v2v edges + self loops
#define FDIM     64                // hidden*heads == out_dim == 64 everywhere

typedef __attribute__((ext_vector_type(2))) float v2f;
typedef __attribute__((ext_vector_type(8))) float v8f;

// ---------------------------------------------------------------- utilities

__device__ __forceinline__ float waveSum(float v) {
#pragma unroll
    for (int m = 16; m >= 1; m >>= 1) v += __shfl_xor(v, m, 32);
    return v;
}

__global__ void fill_kernel(float* __restrict__ p, float v, int n) {
    int i = blockIdx.x * blockDim.x + threadIdx.x;
    if (i < n) p[i] = v;
}

// sum of ea_v2v columns (for self-loop fill_value='mean')
__global__ void easum_kernel(const float* __restrict__ ea, int E, float* __restrict__ out) {
    float a = 0.f, b = 0.f;
    for (int i = blockIdx.x * blockDim.x + threadIdx.x; i < E; i += gridDim.x * blockDim.x) {
        a += ea[2 * i];
        b += ea[2 * i + 1];
    }
    a = waveSum(a);
    b = waveSum(b);
    if ((threadIdx.x & 31) == 0) {
        atomicAdd(&out[0], a);
        atomicAdd(&out[1], b);
    }
}

// ---------------------------------------------------------------- WMMA GEMM
// C[M,64] = A[M,K] @ B[K,64] + bias[64], fp32 via V_WMMA_F32_16X16X4_F32.
// One wave computes a 16x64 tile: 4 accumulators, K fully unrolled (K in {8,16,64}).
// A loads: one b64 per K-step from a row-clamped address + select-to-zero
// (no exec-mask divergence in the WMMA loop; EXEC stays all-ones).
// Store epilogue: wave-uniform full-tile fast path (unguarded store clause);
// guarded path only for the final partial tile.
template <int K>
__global__ void wmma_gemm_bias(const float* __restrict__ A,
                               const float* __restrict__ B,
                               const float* __restrict__ bias,
                               float* __restrict__ C,
                               int M) {
    int wave = (blockIdx.x * blockDim.x + threadIdx.x) >> 5;
    int lane = threadIdx.x & 31;
    int m0 = wave * 16;
    if (m0 >= M) return;                  // wave-uniform

    int half = (lane >= 16) ? 1 : 0;      // lanes 16-31 carry K+2,K+3 (A) / rows M+8 (C)
    int mr   = lane & 15;
    int row  = m0 + mr;
    bool rowOK = (row < M);
    int rowc = rowOK ? row : (M - 1);     // clamp: always load valid memory

    v8f acc[4];
#pragma unroll
    for (int n = 0; n < 4; ++n)
#pragma unroll
        for (int i = 0; i < 8; ++i) acc[n][i] = 0.f;

    const float* Arow = A + (size_t)rowc * K;

#pragma unroll
    for (int k = 0; k < K; k += 4) {
        v2f a = *(const v2f*)(Arow + k + 2 * half);   // global_load_b64
        if (!rowOK) { a.x = 0.f; a.y = 0.f; }         // v_cndmask, no branch
#pragma unroll
        for (int n = 0; n < 4; ++n) {
            int col = n * 16 + mr;
            v2f b;
            b.x = B[(size_t)(k + 2 * half)     * FDIM + col];
            b.y = B[(size_t)(k + 2 * half + 1) * FDIM + col];
            acc[n] = __builtin_amdgcn_wmma_f32_16x16x4_f32(
                false, a, false, b, (short)0, acc[n], false, false);
        }
    }

    float* Cbase = C + (size_t)(m0 + 8 * half) * FDIM + mr;  // row m0+8*half, col mr
    if (m0 + 16 <= M) {
        // full tile: unguarded store clause
#pragma unroll
        for (int n = 0; n < 4; ++n) {
            float bv = bias[n * 16 + mr];
#pragma unroll
            for (int r = 0; r < 8; ++r)
                Cbase[(size_t)r * FDIM + n * 16] = acc[n][r] + bv;
        }
    } else {
        // tail tile: per-row guard
#pragma unroll
        for (int n = 0; n < 4; ++n) {
            float bv = bias[n * 16 + mr];
#pragma unroll
            for (int r = 0; r < 8; ++r) {
                int rr = m0 + r + 8 * half;
                if (rr < M) C[(size_t)rr * FDIM + n * 16 + mr] = acc[n][r] + bv;
            }
        }
    }
}

// ---------------------------------------------------------------- edge passes
// One wave per edge; lane handles hc=lane and hc=lane+32 of the 64-wide rows.
// att is flat[64] with hc == h*C + c in both layer configs.

template <int NHEADS>
__global__ void edge_logits_kernel(const float* __restrict__ xl,
                                   const float* __restrict__ xr,
                                   const int* __restrict__ srcI,
                                   const int* __restrict__ dstI,
                                   const float* __restrict__ ea,
                                   const float* __restrict__ We,   // [2,64]
                                   const float* __restrict__ att,  // [64]
                                   const float* __restrict__ eaSum,
                                   float eaScale,
                                   int Etot, int Ereal,
                                   float* __restrict__ logits,
                                   float* __restrict__ mmax) {
    int e = (blockIdx.x * blockDim.x + threadIdx.x) >> 5;
    if (e >= Etot) return;
    int lane = threadIdx.x & 31;

    int src, dst;
    float ea0, ea1;
    if (e < Ereal) {
        src = srcI[e]; dst = dstI[e];
        ea0 = ea[2 * e]; ea1 = ea[2 * e + 1];
    } else {  // self loop, edge attr = mean
        src = dst = e - Ereal;
        ea0 = eaSum[0] * eaScale;
        ea1 = eaSum[1] * eaScale;
    }

    float p0, p1;
    {
        int hc = lane;
        float z = xl[(size_t)src * FDIM + hc] + xr[(size_t)dst * FDIM + hc]
                + ea0 * We[hc] + ea1 * We[FDIM + hc];
        z = z > 0.f ? z : 0.2f * z;          // leaky_relu(0.2)
        p0 = z * att[hc];
    }
    {
        int hc = lane + 32;
        float z = xl[(size_t)src * FDIM + hc] + xr[(size_t)dst * FDIM + hc]
                + ea0 * We[hc] + ea1 * We[FDIM + hc];
        z = z > 0.f ? z : 0.2f * z;
        p1 = z * att[hc];
    }

    if (NHEADS == 2) {
        float s0 = waveSum(p0);
        float s1 = waveSum(p1);
        if (lane == 0) {
            logits[2 * e]     = s0;
            logits[2 * e + 1] = s1;
            atomicMax(&mmax[2 * dst],     s0);  // float atomic max -> global_atomic_max_num_f32
            atomicMax(&mmax[2 * dst + 1], s1);
        }
    } else {
        float s = waveSum(p0 + p1);
        if (lane == 0) {
            logits[e] = s;
            atomicMax(&mmax[dst], s);
        }
    }
}

template <int NHEADS>
__global__ void edge_exp_kernel(const int* __restrict__ dstI,
                                int Etot, int Ereal,
                                float* __restrict__ logits,
                                const float* __restrict__ mmax,
                                float* __restrict__ den) {
    int e = blockIdx.x * blockDim.x + threadIdx.x;
    if (e >= Etot) return;
    int dst = (e < Ereal) ? dstI[e] : (e - Ereal);
#pragma unroll
    for (int h = 0; h < NHEADS; ++h) {
        float ex = expf(logits[e * NHEADS + h] - mmax[dst * NHEADS + h]);
        logits[e * NHEADS + h] = ex;            // reuse buffer: now holds exp
        atomicAdd(&den[dst * NHEADS + h], ex);
    }
}

template <int NHEADS>
__global__ void edge_scatter_kernel(const float* __restrict__ xl,
                                    const int* __restrict__ srcI,
                                    const int* __restrict__ dstI,
                                    int Etot, int Ereal,
                                    const float* __restrict__ exv,
                                    const float* __restrict__ den,
                                    float* __restrict__ out) {
    int e = (blockIdx.x * blockDim.x + threadIdx.x) >> 5;
    if (e >= Etot) return;
    int lane = threadIdx.x & 31;
    int src, dst;
    if (e < Ereal) { src = srcI[e]; dst = dstI[e]; }
    else           { src = dst = e - Ereal; }

    float a0, a1;
    if (NHEADS == 2) {
        a0 = exv[2 * e]     / den[2 * dst];
        a1 = exv[2 * e + 1] / den[2 * dst + 1];
    } else {
        a0 = a1 = exv[e] / den[dst];
    }
    atomicAdd(&out[(size_t)dst * FDIM + lane],      xl[(size_t)src * FDIM + lane]      * a0);
    atomicAdd(&out[(size_t)dst * FDIM + lane + 32], xl[(size_t)src * FDIM + lane + 32] * a1);
}

// ---------------------------------------------------------------- epilogues

// out = elu(acc + b1 [+ b2]), elementwise over N*64
__global__ void elu_bias_kernel(const float* __restrict__ acc,
                                const float* __restrict__ b1,
                                const float* __restrict__ b2,
                                float* __restrict__ out, int n) {
    int i = blockIdx.x * blockDim.x + threadIdx.x;
    if (i >= n) return;
    int c = i & (FDIM - 1);
    float x = acc[i] + b1[c] + (b2 ? b2[c] : 0.f);
    out[i] = x > 0.f ? x : expm1f(x);
}

// layernorm over last dim 64; one wave per node, 2 elements per lane
__global__ void ln_bias_kernel(const float* __restrict__ acc,
                               const float* __restrict__ b1,
                               const float* __restrict__ b2,
                               const float* __restrict__ g,
                               const float* __restrict__ b,
                               float* __restrict__ out, int N) {
    int node = (blockIdx.x * blockDim.x + threadIdx.x) >> 5;
    if (node >= N) return;
    int lane = threadIdx.x & 31;
    size_t base = (size_t)node * FDIM;
    float x0 = acc[base + lane]      + b1[lane]      + (b2 ? b2[lane]      : 0.f);
    float x1 = acc[base + lane + 32] + b1[lane + 32] + (b2 ? b2[lane + 32] : 0.f);
    float mu = waveSum(x0 + x1) * (1.f / FDIM);
    float d0 = x0 - mu, d1 = x1 - mu;
    float var = waveSum(d0 * d0 + d1 * d1) * (1.f / FDIM);
    float inv = rsqrtf(var + 1e-5f);
    out[base + lane]      = d0 * inv * g[lane]      + b[lane];
    out[base + lane + 32] = d1 * inv * g[lane + 32] + b[lane + 32];
}

// ---------------------------------------------------------------- host

struct ConvP { const float *Wl, *bl, *Wr, *br, *We, *att, *bias; };

extern "C" void kernel_launch(void* const* d_in, const int* in_sizes, int n_in,
                              void* d_out, int out_size, void* d_ws, size_t ws_size,
                              hipStream_t stream) {
    const float* xVeh  = (const float*)d_in[0];
    const float* xRsu  = (const float*)d_in[1];
    const int*   eiV2V = (const int*)d_in[2];       // [2, E_V2V]
    const float* eaV2V = (const float*)d_in[3];
    const int*   v2iS  = (const int*)d_in[4];
    const int*   v2iD  = (const int*)d_in[5];
    const float* eaV2I = (const float*)d_in[6];
    const int*   i2vS  = (const int*)d_in[7];
    const int*   i2vD  = (const int*)d_in[8];
    const float* eaI2V = (const float*)d_in[9];

    auto conv = [&](int base) {
        ConvP c;
        c.Wl   = (const float*)d_in[base + 0];
        c.bl   = (const float*)d_in[base + 1];
        c.Wr   = (const float*)d_in[base + 2];
        c.br   = (const float*)d_in[base + 3];
        c.We   = (const float*)d_in[base + 4];
        c.att  = (const float*)d_in[base + 5];
        c.bias = (const float*)d_in[base + 6];
        return c;
    };
    ConvP c0v2v = conv(10), c0v2i = conv(17), c0i2v = conv(24);
    ConvP c1v2v = conv(31), c1v2i = conv(38), c1i2v = conv(45);
    const float* lnVg = (const float*)d_in[52];
    const float* lnVb = (const float*)d_in[53];
    const float* lnRg = (const float*)d_in[54];
    const float* lnRb = (const float*)d_in[55];

    // workspace layout (floats) -- all tables are small enough to live in L2
    float* ws = (float*)d_ws;
    size_t o = 0;
    float* eaSum = ws + o; o += 16;
    float* xlV1  = ws + o; o += (size_t)N_VEH * FDIM;  // v2v lin_l(vehicles)
    float* xrV1  = ws + o; o += (size_t)N_VEH * FDIM;  // v2v lin_r(vehicles)
    float* xlV2  = ws + o; o += (size_t)N_VEH * FDIM;  // v2i lin_l(vehicles)
    float* xrV2  = ws + o; o += (size_t)N_VEH * FDIM;  // i2v lin_r(vehicles)
    float* xlR   = ws + o; o += (size_t)N_RSU * FDIM;  // i2v lin_l(rsu)
    float* xrR   = ws + o; o += (size_t)N_RSU * FDIM;  // v2i lin_r(rsu)
    float* v1    = ws + o; o += (size_t)N_VEH * FDIM;
    float* r1    = ws + o; o += (size_t)N_RSU * FDIM;
    float* accV  = ws + o; o += (size_t)N_VEH * FDIM;
    float* accR  = ws + o; o += (size_t)N_RSU * FDIM;
    float* elog  = ws + o; o += (size_t)E_V2V_SL * 2;  // logits->exp, reused per conv
    float* mmax  = ws + o; o += (size_t)N_VEH * 2;
    float* dsum  = ws + o; o += (size_t)N_VEH * 2;

    auto gemm = [&](const float* A, const float* B, const float* bias,
                    float* C, int M, int K) {
        int tiles  = (M + 15) / 16;
        int blocks = (tiles * 32 + 255) / 256;
        if (K == 8)
            wmma_gemm_bias<8><<<blocks, 256, 0, stream>>>(A, B, bias, C, M);
        else if (K == 16)
            wmma_gemm_bias<16><<<blocks, 256, 0, stream>>>(A, B, bias, C, M);
        else
            wmma_gemm_bias<64><<<blocks, 256, 0, stream>>>(A, B, bias, C, M);
    };

    auto run_edges = [&](int nheads, const float* xl, const float* xr,
                         const int* src, const int* dst, const float* ea,
                         const ConvP& p, float eaScale,
                         int Etot, int Ereal, int Ndst, float* outAcc) {
        int nm = Ndst * nheads;
        fill_kernel<<<(nm + 255) / 256, 256, 0, stream>>>(mmax, -INFINITY, nm);
        hipMemsetAsync(dsum, 0, (size_t)nm * sizeof(float), stream);
        int wblocks = ((Etot * 32) + 255) / 256;   // wave per edge
        int tblocks = (Etot + 255) / 256;          // thread per edge
        if (nheads == 2) {
            edge_logits_kernel<2><<<wblocks, 256, 0, stream>>>(
                xl, xr, src, dst, ea, p.We, p.att, eaSum, eaScale, Etot, Ereal, elog, mmax);
            edge_exp_kernel<2><<<tblocks, 256, 0, stream>>>(dst, Etot, Ereal, elog, mmax, dsum);
            edge_scatter_kernel<2><<<wblocks, 256, 0, stream>>>(
                xl, src, dst, Etot, Ereal, elog, dsum, outAcc);
        } else {
            edge_logits_kernel<1><<<wblocks, 256, 0, stream>>>(
                xl, xr, src, dst, ea, p.We, p.att, eaSum, eaScale, Etot, Ereal, elog, mmax);
            edge_exp_kernel<1><<<tblocks, 256, 0, stream>>>(dst, Etot, Ereal, elog, mmax, dsum);
            edge_scatter_kernel<1><<<wblocks, 256, 0, stream>>>(
                xl, src, dst, Etot, Ereal, elog, dsum, outAcc);
        }
    };

    // edge-attr mean for v2v self loops
    hipMemsetAsync(eaSum, 0, 2 * sizeof(float), stream);
    easum_kernel<<<256, 256, 0, stream>>>(eaV2V, E_V2V, eaSum);

    // ----- layer 0: node lifts (WMMA fp32 GEMMs) -----
    gemm(xVeh, c0v2v.Wl, c0v2v.bl, xlV1, N_VEH, 16);
    gemm(xVeh, c0v2v.Wr, c0v2v.br, xrV1, N_VEH, 16);
    gemm(xVeh, c0v2i.Wl, c0v2i.bl, xlV2, N_VEH, 16);
    gemm(xRsu, c0v2i.Wr, c0v2i.br, xrR,  N_RSU, 8);
    gemm(xRsu, c0i2v.Wl, c0i2v.bl, xlR,  N_RSU, 8);
    gemm(xVeh, c0i2v.Wr, c0i2v.br, xrV2, N_VEH, 16);

    hipMemsetAsync(accV, 0, (size_t)N_VEH * FDIM * sizeof(float), stream);
    hipMemsetAsync(accR, 0, (size_t)N_RSU * FDIM * sizeof(float), stream);

    run_edges(2, xlV1, xrV1, eiV2V, eiV2V + E_V2V, eaV2V, c0v2v,
              1.f / E_V2V, E_V2V_SL, E_V2V, N_VEH, accV);
    run_edges(2, xlR,  xrV2, i2vS, i2vD, eaI2V, c0i2v, 0.f, E_I2V, E_I2V, N_VEH, accV);
    run_edges(2, xlV2, xrR,  v2iS, v2iD, eaV2I, c0v2i, 0.f, E_V2I, E_V2I, N_RSU, accR);

    elu_bias_kernel<<<(N_VEH * FDIM + 255) / 256, 256, 0, stream>>>(
        accV, c0v2v.bias, c0i2v.bias, v1, N_VEH * FDIM);
    elu_bias_kernel<<<(N_RSU * FDIM + 255) / 256, 256, 0, stream>>>(
        accR, c0v2i.bias, nullptr, r1, N_RSU * FDIM);

    // ----- layer 1: node lifts -----
    gemm(v1, c1v2v.Wl, c1v2v.bl, xlV1, N_VEH, 64);
    gemm(v1, c1v2v.Wr, c1v2v.br, xrV1, N_VEH, 64);
    gemm(v1, c1v2i.Wl, c1v2i.bl, xlV2, N_VEH, 64);
    gemm(r1, c1v2i.Wr, c1v2i.br, xrR,  N_RSU, 64);
    gemm(r1, c1i2v.Wl, c1i2v.bl, xlR,  N_RSU, 64);
    gemm(v1, c1i2v.Wr, c1i2v.br, xrV2, N_VEH, 64);

    hipMemsetAsync(accV, 0, (size_t)N_VEH * FDIM * sizeof(float), stream);
    hipMemsetAsync(accR, 0, (size_t)N_RSU * FDIM * sizeof(float), stream);

    run_edges(1, xlV1, xrV1, eiV2V, eiV2V + E_V2V, eaV2V, c1v2v,
              1.f / E_V2V, E_V2V_SL, E_V2V, N_VEH, accV);
    run_edges(1, xlR,  xrV2, i2vS, i2vD, eaI2V, c1i2v, 0.f, E_I2V, E_I2V, N_VEH, accV);
    run_edges(1, xlV2, xrR,  v2iS, v2iD, eaV2I, c1v2i, 0.f, E_V2I, E_V2I, N_RSU, accR);

    // ----- epilogue: bias sum + layernorm, write straight to d_out -----
    float* outVeh = (float*)d_out;
    float* outRsu = outVeh + (size_t)N_VEH * FDIM;
    ln_bias_kernel<<<(N_VEH * 32 + 255) / 256, 256, 0, stream>>>(
        accV, c1v2v.bias, c1i2v.bias, lnVg, lnVb, outVeh, N_VEH);
    ln_bias_kernel<<<(N_RSU * 32 + 255) / 256, 256, 0, stream>>>(
        accR, c1v2i.bias, nullptr, lnRg, lnRb, outRsu, N_RSU);
}